// Edge_Discriminator_72859825210006
// MI455X (gfx1250) — compile-verified
//
#include <hip/hip_runtime.h>
#include <hip/hip_bf16.h>

// ---- reference constants ----
#define EOS        1e-10f
#define ALPHA      0.5f
#define BIAS_C     0.0001f
#define N_NODES    8192
#define N_EDGES    262144
#define INPUT_DIM  512
#define HIDDEN     128

typedef float v2f __attribute__((ext_vector_type(2)));
typedef float v4f __attribute__((ext_vector_type(4)));
typedef float v8f __attribute__((ext_vector_type(8)));

// ============================================================================
// Kernel 1: h = relu(F @ W^T + b)   [8192x128, K=512], fp32 WMMA 16x16x4.
// One wave per 16x16 output tile; A = F tile (16x4), B = W^T tile (4x16).
// A layout: lane L holds row M=L&15, K = 2*(L>>4) + {0,1} in 2 VGPRs.
// B layout (mirror, N across lanes): lane L holds col N=L&15, K = 2*(L>>4)+{0,1}.
// C layout: VGPR r -> row r + 8*(L>>4), col L&15.
// ============================================================================
__global__ void gemm_relu_kernel(const float* __restrict__ F,
                                 const float* __restrict__ W,
                                 const float* __restrict__ b,
                                 float* __restrict__ h) {
    const int wave = (blockIdx.x * blockDim.x + threadIdx.x) >> 5; // 0..4095
    const int lane = threadIdx.x & 31;
    const int mtile = wave >> 3;   // 512 row tiles
    const int ntile = wave & 7;    // 8 col tiles
    const int m0 = mtile * 16;
    const int n0 = ntile * 16;

    const int row = lane & 15;
    const int kb  = (lane >> 4) << 1;   // 0 or 2
    const float* __restrict__ Fp = F + (size_t)(m0 + row) * INPUT_DIM + kb;
    const float* __restrict__ Wp = W + (size_t)(n0 + row) * INPUT_DIM + kb;

    v8f acc = {};
    #pragma unroll 4
    for (int k = 0; k < INPUT_DIM; k += 4) {
        v2f a  = *(const v2f*)(Fp + k);
        v2f bb = *(const v2f*)(Wp + k);
        // 8 args: (neg_a, A, neg_b, B, c_mod, C, reuse_a, reuse_b)
        acc = __builtin_amdgcn_wmma_f32_16x16x4_f32(
            false, a, false, bb, (short)0, acc, false, false);
    }

    const int cn   = n0 + (lane & 15);
    const int mb   = m0 + ((lane >> 4) << 3);
    const float bn = b[cn];
    #pragma unroll
    for (int r = 0; r < 8; ++r) {
        float v = acc[r] + bn;
        h[(size_t)(mb + r) * HIDDEN + cn] = v > 0.0f ? v : 0.0f;
    }
}

// ============================================================================
// Kernel 2: deg_lp/deg_hp = 1.0 (identity's row-sum contribution)
// ============================================================================
__global__ void init_deg_kernel(float* __restrict__ deg_lp,
                                float* __restrict__ deg_hp) {
    int i = blockIdx.x * blockDim.x + threadIdx.x;
    if (i < N_NODES) { deg_lp[i] = 1.0f; deg_hp[i] = 1.0f; }
}

// ============================================================================
// Kernel 3: per-edge gate.  raw = (h[e0]+h[e1]) . Wc + b_edge  with
// Wc = 0.5*(Wa+Wb)  (algebraic collapse of 0.5*(s1+s2)).
// Writes weights_lp/hp to d_out and accumulates row degrees atomically.
// h (4 MB) is L2-resident on 192 MB L2, so the gathers are cheap.
// ============================================================================
__global__ void edge_gate_kernel(const float* __restrict__ h,
                                 const float* __restrict__ W_edge,
                                 const float* __restrict__ b_edge,
                                 const float* __restrict__ noise,
                                 const int* __restrict__ e0,
                                 const int* __restrict__ e1,
                                 float* __restrict__ wlp_out,
                                 float* __restrict__ whp_out,
                                 float* __restrict__ deg_lp,
                                 float* __restrict__ deg_hp) {
    __shared__ float sWc[HIDDEN];
    __shared__ float sB;
    const int t = threadIdx.x;
    if (t < HIDDEN) sWc[t] = 0.5f * (W_edge[t] + W_edge[HIDDEN + t]);
    if (t == 0)     sB = b_edge[0];
    __syncthreads();

    const int e = blockIdx.x * blockDim.x + t;   // E is a multiple of 256
    const int i = e0[e];
    const int j = e1[e];

    const v4f* __restrict__ hi = (const v4f*)(h + (size_t)i * HIDDEN);
    const v4f* __restrict__ hj = (const v4f*)(h + (size_t)j * HIDDEN);
    const v4f* __restrict__ wc = (const v4f*)sWc;

    float raw = sB;
    #pragma unroll 8
    for (int q = 0; q < HIDDEN / 4; ++q) {
        v4f a = hi[q], c = hj[q], w = wc[q];
        raw += (a.x + c.x) * w.x + (a.y + c.y) * w.y
             + (a.z + c.z) * w.z + (a.w + c.w) * w.w;
    }

    const float nz   = noise[e];
    const float eps  = (2.0f * BIAS_C - 1.0f) * nz + (1.0f - BIAS_C);
    const float gate = logf(eps) - log1pf(-eps) + raw;  // TEMPERATURE == 1
    const float wlp  = 1.0f / (1.0f + expf(-gate));
    const float whp  = 1.0f - wlp;

    wlp_out[e] = wlp;
    whp_out[e] = whp;
    atomicAdd(&deg_lp[i], wlp);
    atomicAdd(&deg_hp[i], whp);
}

// ============================================================================
// Kernel 4: dinv = 1/(sqrt(deg)+EOS)
// ============================================================================
__global__ void dinv_kernel(const float* __restrict__ deg_lp,
                            const float* __restrict__ deg_hp,
                            float* __restrict__ dinv_lp,
                            float* __restrict__ dinv_hp) {
    int i = blockIdx.x * blockDim.x + threadIdx.x;
    if (i < N_NODES) {
        dinv_lp[i] = 1.0f / (sqrtf(deg_lp[i]) + EOS);
        dinv_hp[i] = 1.0f / (sqrtf(deg_hp[i]) + EOS);
    }
}

// ============================================================================
// Kernel 5: base fill of both dense matrices — THE bandwidth bottleneck
// (536 MB of writes -> ~23 us floor at 23.3 TB/s).  Non-temporal 128-bit
// stores: write-once data, keep L2 free for h / degree vectors.
//   adj_lp base: diag = dinv_lp[i]^2, else 0
//   adj_hp base: identity
// ============================================================================
__global__ void fill_adj_kernel(float* __restrict__ adj_lp,
                                float* __restrict__ adj_hp,
                                const float* __restrict__ dinv_lp) {
    const size_t total  = (size_t)N_NODES * (N_NODES / 4);   // v4f per matrix
    const size_t stride = (size_t)gridDim.x * blockDim.x;
    for (size_t v = (size_t)blockIdx.x * blockDim.x + threadIdx.x;
         v < total; v += stride) {
        const int i  = (int)(v >> 11);       // 8192/4 = 2048 v4f per row
        const int jb = (int)(v & 2047);
        v4f zl = {0.0f, 0.0f, 0.0f, 0.0f};
        v4f zh = {0.0f, 0.0f, 0.0f, 0.0f};
        if (jb == (i >> 2)) {                // this vec contains the diagonal
            const float d = dinv_lp[i];
            const int sub = i & 3;
            zl[sub] = d * d;
            zh[sub] = 1.0f;
        }
        __builtin_nontemporal_store(zl, (v4f*)adj_lp + v);
        __builtin_nontemporal_store(zh, (v4f*)adj_hp + v);
    }
}

// ============================================================================
// Kernel 6: sparse scatter of the edge entries on top of the base pattern.
//   adj_lp[i,j] = (wlp + [i==j]) * dinv_lp[i]*dinv_lp[j]
//   adj_hp[i,j] = [i==j] - (whp + [i==j]) * dinv_hp[i]*dinv_hp[j] * ALPHA
// Plain stores reproduce JAX .set scatter semantics (one dup wins).
// ============================================================================
__global__ void scatter_edges_kernel(const int* __restrict__ e0,
                                     const int* __restrict__ e1,
                                     const float* __restrict__ wlp,
                                     const float* __restrict__ whp,
                                     const float* __restrict__ dinv_lp,
                                     const float* __restrict__ dinv_hp,
                                     float* __restrict__ adj_lp,
                                     float* __restrict__ adj_hp) {
    const int e = blockIdx.x * blockDim.x + threadIdx.x;
    const int i = e0[e];
    const int j = e1[e];
    const float self = (i == j) ? 1.0f : 0.0f;
    const size_t off = ((size_t)i << 13) + (size_t)j;
    const float dl = dinv_lp[i] * dinv_lp[j];
    const float dh = dinv_hp[i] * dinv_hp[j];
    adj_lp[off] = (wlp[e] + self) * dl;
    adj_hp[off] = self - (whp[e] + self) * dh * ALPHA;
}

// ============================================================================
// launch
// ============================================================================
extern "C" void kernel_launch(void* const* d_in, const int* in_sizes, int n_in,
                              void* d_out, int out_size, void* d_ws, size_t ws_size,
                              hipStream_t stream) {
    (void)in_sizes; (void)n_in; (void)out_size; (void)ws_size;

    const float* features = (const float*)d_in[0];   // [8192, 512]
    const float* W_emb    = (const float*)d_in[1];   // [128, 512]
    const float* b_emb    = (const float*)d_in[2];   // [128]
    const float* W_edge   = (const float*)d_in[3];   // [1, 256]
    const float* b_edge   = (const float*)d_in[4];   // [1]
    const float* noise    = (const float*)d_in[5];   // [262144]
    const int*   edges    = (const int*)d_in[6];     // [2, 262144]
    const int*   e0 = edges;
    const int*   e1 = edges + N_EDGES;

    // d_out layout: adj_lp | adj_hp | weights_lp | weights_hp
    float* out     = (float*)d_out;
    float* adj_lp  = out;
    float* adj_hp  = out + (size_t)N_NODES * N_NODES;
    float* wlp_out = out + 2ull * N_NODES * N_NODES;
    float* whp_out = wlp_out + N_EDGES;

    // workspace: h (4 MB) + degree / dinv vectors
    float* h       = (float*)d_ws;
    float* deg_lp  = h + (size_t)N_NODES * HIDDEN;
    float* deg_hp  = deg_lp + N_NODES;
    float* dinv_lp = deg_hp + N_NODES;
    float* dinv_hp = dinv_lp + N_NODES;

    // 1) embedding GEMM + ReLU : 4096 16x16 tiles, 8 waves/block
    gemm_relu_kernel<<<512, 256, 0, stream>>>(features, W_emb, b_emb, h);

    // 2) degree init (identity contribution)
    init_deg_kernel<<<(N_NODES + 255) / 256, 256, 0, stream>>>(deg_lp, deg_hp);

    // 3) edge gating + weights + degree accumulation
    edge_gate_kernel<<<N_EDGES / 256, 256, 0, stream>>>(
        h, W_edge, b_edge, noise, e0, e1, wlp_out, whp_out, deg_lp, deg_hp);

    // 4) inverse sqrt degrees
    dinv_kernel<<<(N_NODES + 255) / 256, 256, 0, stream>>>(
        deg_lp, deg_hp, dinv_lp, dinv_hp);

    // 5) dense base fill (bandwidth-bound, NT stores)
    fill_adj_kernel<<<8192, 256, 0, stream>>>(adj_lp, adj_hp, dinv_lp);

    // 6) sparse edge scatter
    scatter_edges_kernel<<<N_EDGES / 256, 256, 0, stream>>>(
        e0, e1, wlp_out, whp_out, dinv_lp, dinv_hp, adj_lp, adj_hp);
}